// DeepBDC_31061203484793
// MI455X (gfx1250) — compile-verified
//
#include <hip/hip_runtime.h>
#include <hip/hip_bf16.h>

// ---------------------------------------------------------------------------
// DeepBDC forward for MI455X (gfx1250, wave32, WMMA + TDM).
//
//   K1 bdc_support_tv  : 800 blocks  -> tv[800][2080] in ws       (TDM + WMMA)
//   K2 support_mean    : prototypes sup[10][2080] = mean over 80
//   K3 support_norm    : s2[10] = ||sup_c||^2
//   K4 bdc_query_score : 16000 blocks -> fused dcov + dot -> score (TDM + WMMA)
//   K5 loss_kernel     : log-softmax NLL mean -> d_out[160000]
// ---------------------------------------------------------------------------

#define DIMX    64
#define HW      49
#define XSTRIDE 49          // TDM writes contiguous 49-float rows into LDS
#define GSTRIDE 65          // padded LDS row stride for G (avoid bank conflicts)
#define NTRIU   2080        // 64*65/2
#define NWAY    10
#define NSUP    800
#define NQRY    16000
#define SUPPG   80          // support samples per class

typedef __attribute__((ext_vector_type(2))) float        v2f;
typedef __attribute__((ext_vector_type(8))) float        v8f;
typedef __attribute__((ext_vector_type(4))) unsigned int v4u;
typedef __attribute__((ext_vector_type(4))) int          v4i;
typedef __attribute__((ext_vector_type(8))) int          v8i;

struct BdcShared {
    float X[DIMX * XSTRIDE];   // 3136 f : 64x49 input slab (TDM destination)
    float G[DIMX * GSTRIDE];   // 4160 f : gram -> dcov (in place)
    float dg[DIMX];            // diagonal of gram
    float rm[DIMX];            // row means of dcov (== col means: symmetric)
    float red[DIMX * 4];       // 256 f  : reduction scratch (reused)
    float tm;                  // total mean
};

// ---- Tensor Data Mover: load one 64x49 fp32 tile (row stride 49) into LDS --
// D# descriptor per CDNA5 ISA. 2D tensor:
//   tensor_dim0 = 49 (row length), tensor_dim1 = 64 (rows)
//   tile_dim0   = 49, tile_dim1   = 64, tensor_dim0_stride = 49, data_size=4B
__device__ __forceinline__ void tdm_load_x(const float* __restrict__ gptr,
                                           float* lds_ptr) {
    // LDS aperture: flat addr = {SHARED_BASE[15:0],16'b0} | offset, so the
    // low 32 bits of the generic address are the LDS byte offset.
    const unsigned lds_addr = (unsigned)(unsigned long long)lds_ptr;
    const unsigned long long ga = (unsigned long long)gptr;

    v4u g0;
    g0[0] = 1u;                                       // count=1, user mode
    g0[1] = lds_addr;                                 // lds_addr [63:32]
    g0[2] = (unsigned)(ga & 0xFFFFFFFFu);             // global_addr lo32
    g0[3] = (unsigned)((ga >> 32) & 0x01FFFFFFu)      // global_addr [120:64]
          | (2u << 30);                               // type=2 ("image")

    v8i g1;
    g1[0] = (2 << 16);        // wg_mask=0, data_size=2 (4 bytes), no pad/iter
    g1[1] = (HW << 16);       // tensor_dim0 lo16 = 49
    g1[2] = (DIMX << 16);     // tensor_dim0 hi16=0 | tensor_dim1 lo16 = 64
    g1[3] = (HW << 16);       // tensor_dim1 hi16=0 | tile_dim0 = 49
    g1[4] = DIMX;             // tile_dim1 = 64, tile_dim2 = 0
    g1[5] = HW;               // tensor_dim0_stride lo32 = 49
    g1[6] = 0;                // stride hi16 = 0, tensor_dim1_stride lo16 = 0
    g1[7] = 0;

    v4i gz = {0, 0, 0, 0};    // groups 2/3 unused (2D tile)
#if defined(__clang_major__) && (__clang_major__ >= 23)
    v8i gz8 = {0, 0, 0, 0, 0, 0, 0, 0};
    __builtin_amdgcn_tensor_load_to_lds(g0, g1, gz, gz, gz8, 0);
#else
    __builtin_amdgcn_tensor_load_to_lds(g0, g1, gz, gz, 0);
#endif
}

// Compute centered-ready state for one sample: after return,
//   s.G holds dcov, s.rm row means, s.tm total mean, so
//   v[i][j] = s.G[i*GSTRIDE+j] - s.rm[i] - s.rm[j] + s.tm.
// blockDim.x must be 256 (8 wave32 waves).
__device__ __forceinline__ void bdc_dcov(const float* __restrict__ feat_b,
                                         float expt, BdcShared& s) {
    const int t = threadIdx.x;

    // ---- stage X into LDS via the Tensor Data Mover (wave 0 issues) ----
    if (t < 32) {
        tdm_load_x(feat_b, &s.X[0]);
        __builtin_amdgcn_s_wait_tensorcnt(0);
    }
    __syncthreads();

    // ---- Gram G = X * X^T via V_WMMA_F32_16X16X4_F32 ----
    const int lane  = t & 31;
    const int wave  = t >> 5;
    const int half  = lane >> 4;    // 0: lanes 0-15, 1: lanes 16-31
    const int idx16 = lane & 15;

    #pragma unroll
    for (int tile = 0; tile < 2; ++tile) {
        const int tt = wave * 2 + tile;         // 16 tiles over 8 waves
        const int mt = (tt >> 2) * 16;
        const int nt = (tt & 3) * 16;
        const float* __restrict__ arow = &s.X[(mt + idx16) * XSTRIDE];
        const float* __restrict__ brow = &s.X[(nt + idx16) * XSTRIDE];
        v8f c = {0.f, 0.f, 0.f, 0.f, 0.f, 0.f, 0.f, 0.f};
        // 12 full k-steps of 4 (K = 0..47)
        #pragma unroll
        for (int ks = 0; ks < 12; ++ks) {
            // A 16x4 f32 layout: lane<16 -> K={k0,k0+1}, lane>=16 -> K={k0+2,k0+3}
            const int kb = ks * 4 + half * 2;
            v2f a; a.x = arow[kb]; a.y = arow[kb + 1];
            v2f b; b.x = brow[kb]; b.y = brow[kb + 1];
            c = __builtin_amdgcn_wmma_f32_16x16x4_f32(
                    false, a, false, b, (short)0, c, false, false);
        }
        // remainder k-step: K=48 valid, K=49..51 are zeros (register-masked)
        {
            v2f a; a.x = half ? 0.f : arow[48]; a.y = 0.f;
            v2f b; b.x = half ? 0.f : brow[48]; b.y = 0.f;
            c = __builtin_amdgcn_wmma_f32_16x16x4_f32(
                    false, a, false, b, (short)0, c, false, false);
        }
        // C/D layout: VGPR v, lanes0-15 -> M=v, lanes16-31 -> M=v+8; N=lane&15
        #pragma unroll
        for (int v = 0; v < 8; ++v) {
            s.G[(mt + v + half * 8) * GSTRIDE + nt + idx16] = c[v];
        }
    }
    __syncthreads();

    // ---- diagonal ----
    if (t < DIMX) s.dg[t] = s.G[t * GSTRIDE + t];
    __syncthreads();

    // ---- dcov = sqrt(expt * clip(dg_i + dg_j - 2 g, 0) + 1e-5), in place ----
    {
        const int i  = t >> 2;      // row
        const int jq = t & 3;       // quarter of the row
        const float dgi = s.dg[i];
        float part = 0.f;
        #pragma unroll
        for (int jj = 0; jj < 16; ++jj) {
            const int j = jq * 16 + jj;
            const float g  = s.G[i * GSTRIDE + j];
            const float d2 = dgi + s.dg[j] - 2.0f * g;
            const float d  = sqrtf(expt * fmaxf(d2, 0.0f) + 1e-5f);
            s.G[i * GSTRIDE + j] = d;
            part += d;
        }
        s.red[t] = part;
    }
    __syncthreads();

    if (t < DIMX) {
        const float rs = s.red[t * 4] + s.red[t * 4 + 1] +
                         s.red[t * 4 + 2] + s.red[t * 4 + 3];
        s.rm[t] = rs * (1.0f / DIMX);
    }
    __syncthreads();
    if (t == 0) {
        float tot = 0.f;
        #pragma unroll
        for (int i = 0; i < DIMX; ++i) tot += s.rm[i];
        s.tm = tot * (1.0f / DIMX);
    }
    __syncthreads();
}

// ---- K1: support samples -> tv rows ---------------------------------------
__global__ void __launch_bounds__(256)
bdc_support_tv(const float* __restrict__ feat, const float* __restrict__ temp,
               float* __restrict__ tv) {
    __shared__ BdcShared s;
    const int b = blockIdx.x;                       // 0..799
    if (b + 1 < NSUP)
        __builtin_prefetch(feat + (size_t)(b + 1) * DIMX * HW, 0, 3);
    const float expt = expf(temp[0]);
    bdc_dcov(feat + (size_t)b * DIMX * HW, expt, s);

    float* __restrict__ out = tv + (size_t)b * NTRIU;
    const int t  = threadIdx.x;
    const int i  = t & 63;
    const int jo = t >> 6;                          // 0..3
    const int rowoff = i * DIMX - (i * (i - 1)) / 2;
    const float rmi = s.rm[i];
    const float tm  = s.tm;
    for (int j = i + jo; j < DIMX; j += 4) {
        const float v = s.G[i * GSTRIDE + j] - rmi - s.rm[j] + tm;
        out[rowoff + (j - i)] = v;
    }
}

// ---- K2: class prototypes = mean over 80 support tv rows ------------------
__global__ void __launch_bounds__(256)
support_mean(const float* __restrict__ tv, float* __restrict__ sup) {
    const int e = blockIdx.x * 256 + threadIdx.x;
    if (e >= NWAY * NTRIU) return;
    const int c = e / NTRIU;
    const int k = e - c * NTRIU;
    const float* __restrict__ p = tv + (size_t)(c * SUPPG) * NTRIU + k;
    float acc = 0.f;
    for (int si = 0; si < SUPPG; ++si) acc += p[(size_t)si * NTRIU];
    sup[e] = acc * (1.0f / SUPPG);
}

// ---- K3: prototype squared norms ------------------------------------------
__global__ void __launch_bounds__(256)
support_norm(const float* __restrict__ sup, float* __restrict__ s2) {
    __shared__ float red[8];
    const int c = blockIdx.x;
    float acc = 0.f;
    for (int k = threadIdx.x; k < NTRIU; k += 256) {
        const float v = sup[c * NTRIU + k];
        acc += v * v;
    }
    for (int off = 16; off; off >>= 1) acc += __shfl_xor(acc, off, 32);
    if ((threadIdx.x & 31) == 0) red[threadIdx.x >> 5] = acc;
    __syncthreads();
    if (threadIdx.x == 0) {
        float tot = 0.f;
        #pragma unroll
        for (int w = 0; w < 8; ++w) tot += red[w];
        s2[c] = tot;
    }
}

// ---- K4: query -> fused dcov + prototype dots -> score row ----------------
__global__ void __launch_bounds__(256)
bdc_query_score(const float* __restrict__ feat, const float* __restrict__ temp,
                const float* __restrict__ sup, const float* __restrict__ s2,
                float* __restrict__ score) {
    __shared__ BdcShared s;
    __shared__ float tot[12];
    const int q = blockIdx.x;                       // 0..15999
    const int b = NSUP + q;
    if (q + 1 < NQRY)
        __builtin_prefetch(feat + (size_t)(b + 1) * DIMX * HW, 0, 3);
    const float expt = expf(temp[0]);
    bdc_dcov(feat + (size_t)b * DIMX * HW, expt, s);

    const int t  = threadIdx.x;
    const int i  = t & 63;
    const int jo = t >> 6;
    const int rowoff = i * DIMX - (i * (i - 1)) / 2;
    const float rmi = s.rm[i];
    const float tm  = s.tm;

    float q2 = 0.f;
    float dot[NWAY];
    #pragma unroll
    for (int c = 0; c < NWAY; ++c) dot[c] = 0.f;

    for (int j = i + jo; j < DIMX; j += 4) {
        const float v = s.G[i * GSTRIDE + j] - rmi - s.rm[j] + tm;
        const int k = rowoff + (j - i);
        q2 += v * v;
        #pragma unroll
        for (int c = 0; c < NWAY; ++c) dot[c] += v * sup[c * NTRIU + k];
    }

    // reduce 11 values across the 8 waves
    for (int off = 16; off; off >>= 1) {
        q2 += __shfl_xor(q2, off, 32);
        #pragma unroll
        for (int c = 0; c < NWAY; ++c) dot[c] += __shfl_xor(dot[c], off, 32);
    }
    const int lane = t & 31, wave = t >> 5;
    if (lane == 0) {
        s.red[wave * 12] = q2;
        #pragma unroll
        for (int c = 0; c < NWAY; ++c) s.red[wave * 12 + 1 + c] = dot[c];
    }
    __syncthreads();
    if (t < 12) {
        float v = 0.f;
        #pragma unroll
        for (int w = 0; w < 8; ++w) v += s.red[w * 12 + t];
        tot[t] = v;
    }
    __syncthreads();
    if (t < NWAY) {
        score[(size_t)q * NWAY + t] = -(tot[0] + s2[t] - 2.0f * tot[1 + t]);
    }
}

// ---- K5: NLL of log-softmax, mean over queries ----------------------------
__global__ void __launch_bounds__(256)
loss_kernel(const float* __restrict__ score, const int* __restrict__ label,
            float* __restrict__ out_loss) {
    __shared__ float red[8];
    float acc = 0.f;
    for (int n = threadIdx.x; n < NQRY; n += 256) {
        const float* __restrict__ row = score + (size_t)n * NWAY;
        float m = row[0];
        #pragma unroll
        for (int c = 1; c < NWAY; ++c) m = fmaxf(m, row[c]);
        float se = 0.f;
        #pragma unroll
        for (int c = 0; c < NWAY; ++c) se += expf(row[c] - m);
        const float lse = m + logf(se);
        const int lab = label[n];
        acc += (lse - row[lab]);
    }
    for (int off = 16; off; off >>= 1) acc += __shfl_xor(acc, off, 32);
    if ((threadIdx.x & 31) == 0) red[threadIdx.x >> 5] = acc;
    __syncthreads();
    if (threadIdx.x == 0) {
        float tot = 0.f;
        #pragma unroll
        for (int w = 0; w < 8; ++w) tot += red[w];
        out_loss[0] = tot * (1.0f / NQRY);
    }
}

// ---------------------------------------------------------------------------
extern "C" void kernel_launch(void* const* d_in, const int* in_sizes, int n_in,
                              void* d_out, int out_size, void* d_ws, size_t ws_size,
                              hipStream_t stream) {
    const float* feat  = (const float*)d_in[0];   // (16800, 64, 7, 7) f32
    const float* temp  = (const float*)d_in[1];   // (1,1) f32  (log temp)
    const int*   label = (const int*)d_in[2];     // (16000,) int

    float* out = (float*)d_out;                   // 160000 score + 1 loss
    float* ws  = (float*)d_ws;
    float* tv  = ws;                              // 800 * 2080
    float* sup = ws + (size_t)NSUP * NTRIU;       // 10 * 2080
    float* s2  = sup + (size_t)NWAY * NTRIU;      // 10

    bdc_support_tv<<<NSUP, 256, 0, stream>>>(feat, temp, tv);
    support_mean<<<(NWAY * NTRIU + 255) / 256, 256, 0, stream>>>(tv, sup);
    support_norm<<<NWAY, 256, 0, stream>>>(sup, s2);
    bdc_query_score<<<NQRY, 256, 0, stream>>>(feat, temp, sup, s2, out);
    loss_kernel<<<1, 256, 0, stream>>>(out, label, out + (size_t)NQRY * NWAY);
}